// GlobalTokenSelector_86088324481677
// MI455X (gfx1250) — compile-verified
//
#include <hip/hip_runtime.h>
#include <hip/hip_bf16.h>
#include <stdint.h>

// Problem constants (B=8, L=4096, D=1024, k_target=2048)
#define B_  8
#define L_  4096
#define D_  1024
#define K_  2048

// GEMM tiling
#define TL   128   // L rows per block
#define TK   128   // K cols per block
#define TD   32    // depth chunk per LDS stage
#define LDST 36    // padded LDS row stride (floats): 16B-aligned rows, conflict-free frags

typedef __attribute__((ext_vector_type(2))) float v2f;
typedef __attribute__((ext_vector_type(8))) float v8f;

__device__ __forceinline__ unsigned int orderedKey(float f) {
  // monotonic float -> unsigned mapping (no NaNs in this problem)
  unsigned int u = __float_as_uint(f);
  return (u & 0x80000000u) ? ~u : (u | 0x80000000u);
}

// CDNA5 async global->LDS DMA (VGPR-bypassing, tracked by ASYNCcnt).
// dst: LDS byte offset (low 32 bits of a generic pointer into __shared__),
// src: 64-bit global address. 16 bytes per lane.
__device__ __forceinline__ void asyncLoad16(unsigned int ldsOff, unsigned long long gaddr) {
  asm volatile("global_load_async_to_lds_b128 %0, %1, off"
               :
               : "v"(ldsOff), "v"(gaddr)
               : "memory");
}

__device__ __forceinline__ void waitAsync0() {
#if __has_builtin(__builtin_amdgcn_s_wait_asynccnt)
  __builtin_amdgcn_s_wait_asynccnt(0);
#else
  asm volatile("s_wait_asynccnt 0" ::: "memory");
#endif
}

// ---------------------------------------------------------------- init
__global__ void tsel_init_kernel(unsigned long long* __restrict__ packed,
                                 float* __restrict__ gsizes) {
  int tid = blockIdx.x * blockDim.x + threadIdx.x;
  if (tid < B_ * L_) packed[tid] = 0xFFFFFFFFFFFFFFFFULL;
  if (tid < B_ * K_) gsizes[tid] = 0.0f;
}

// ------------------------------------------------- s2 + x_selected copy
// one wave per selected row k: copy x[b,k,:] to x_selected and compute sum(x^2)
__global__ __launch_bounds__(256)
void tsel_s2_copy_kernel(const float* __restrict__ x,
                         float* __restrict__ xsel,
                         float* __restrict__ s2) {
  const int gwave = (blockIdx.x * 256 + threadIdx.x) >> 5;  // 0 .. B_*K_-1
  const int lane  = threadIdx.x & 31;
  const int b = gwave / K_;
  const int k = gwave - b * K_;
  const float* __restrict__ src = x    + ((size_t)b * L_ + k) * D_;
  float*       __restrict__ dst = xsel + ((size_t)b * K_ + k) * D_;
  float sum = 0.0f;
#pragma unroll
  for (int i = 0; i < D_ / (32 * 4); ++i) {
    const int off = (i * 32 + lane) * 4;
    const float4 v = *(const float4*)(src + off);
    *(float4*)(dst + off) = v;
    sum += v.x * v.x + v.y * v.y + v.z * v.z + v.w * v.w;
  }
#pragma unroll
  for (int off = 16; off >= 1; off >>= 1) sum += __shfl_xor(sum, off, 32);
  if (lane == 0) s2[b * K_ + k] = sum;
}

// ------------------------------------------- fp32 WMMA GEMM + fused argmin
// block = 256 threads = 8 waves; block tile = TL x TK; wave w owns rows
// [16w, 16w+16) of the tile and all 8 column WMMA tiles. Double-buffered
// async-DMA staging overlaps global traffic with the WMMA stream.
__global__ __launch_bounds__(256)
void tsel_argmin_kernel(const float* __restrict__ x,
                        const float* __restrict__ s2,
                        unsigned long long* __restrict__ packed) {
  __shared__ float As[2][TL][LDST];
  __shared__ float Bs[2][TK][LDST];

  const int b    = blockIdx.z;
  const int l0   = blockIdx.y * TL;
  const int k0   = blockIdx.x * TK;
  const int tid  = threadIdx.x;
  const int wave = tid >> 5;
  const int lane = tid & 31;
  const int c    = lane & 15;   // N column (B/C/D) or M row (A) within 16
  const int half = lane >> 4;   // 0: K={0,1}; 1: K={2,3}
  const int kb   = half * 2;
  const int m0   = wave * 16;

  const float* __restrict__ xb = x + (size_t)b * L_ * D_;

  // issue the async copies for one TD-deep stage into buffer `buf`
  auto stage = [&](int buf, int d0) {
#pragma unroll
    for (int i = 0; i < (TL * TD) / (256 * 4); ++i) {
      const int idx = tid + 256 * i;          // one 16B slot per entry
      const int row = idx >> 3;               // TD/4 = 8 float4 per row
      const int c4  = (idx & 7) << 2;
      asyncLoad16((unsigned int)(uintptr_t)&As[buf][row][c4],
                  (unsigned long long)(uintptr_t)(xb + (size_t)(l0 + row) * D_ + d0 + c4));
      asyncLoad16((unsigned int)(uintptr_t)&Bs[buf][row][c4],
                  (unsigned long long)(uintptr_t)(xb + (size_t)(k0 + row) * D_ + d0 + c4));
    }
  };

  v8f acc[TK / 16];
#pragma unroll
  for (int j = 0; j < TK / 16; ++j)
#pragma unroll
    for (int r = 0; r < 8; ++r) acc[j][r] = 0.0f;

  stage(0, 0);
  waitAsync0();
  __syncthreads();

  int buf = 0;
  for (int d0 = 0; d0 < D_; d0 += TD, buf ^= 1) {
    if (d0 + TD < D_) stage(buf ^ 1, d0 + TD);  // DMA next stage under compute

#pragma unroll
    for (int d = 0; d < TD; d += 4) {
      // gather all fragments first so DS loads batch under one wait,
      // then run 8 independent-accumulator WMMAs back-to-back.
      v2f afrag;
      afrag.x = As[buf][m0 + c][d + kb + 0];
      afrag.y = As[buf][m0 + c][d + kb + 1];
      v2f bfrag[TK / 16];
#pragma unroll
      for (int j = 0; j < TK / 16; ++j) {
        bfrag[j].x = Bs[buf][j * 16 + c][d + kb + 0];
        bfrag[j].y = Bs[buf][j * 16 + c][d + kb + 1];
      }
#pragma unroll
      for (int j = 0; j < TK / 16; ++j) {
        acc[j] = __builtin_amdgcn_wmma_f32_16x16x4_f32(
            /*neg_a=*/false, afrag, /*neg_b=*/false, bfrag[j],
            /*c_mod=*/(short)0, acc[j], /*reuse_a=*/false, /*reuse_b=*/false);
      }
    }

    waitAsync0();     // next-stage DMA complete (no-op on last iteration)
    __syncthreads();  // all waves done reading `buf` before it is overwritten
  }

  // ----- epilogue: score = 0.5*s2_k - dot (same argmin as d2), local argmin
  float bestVal[8];
  int   bestIdx[8];
#pragma unroll
  for (int r = 0; r < 8; ++r) { bestVal[r] = 3.4e38f; bestIdx[r] = 0x7FFFFFFF; }

#pragma unroll
  for (int j = 0; j < TK / 16; ++j) {
    const int col = k0 + j * 16 + c;
    const float h = 0.5f * s2[b * K_ + col];
#pragma unroll
    for (int r = 0; r < 8; ++r) {
      const float v = h - acc[j][r];
      if (v < bestVal[r]) { bestVal[r] = v; bestIdx[r] = col; }  // strict: lowest col wins ties
    }
  }

  // reduce over the 16 column-lanes of each half-wave; tie -> lower index
#pragma unroll
  for (int r = 0; r < 8; ++r) {
    float v  = bestVal[r];
    int   id = bestIdx[r];
#pragma unroll
    for (int off = 8; off >= 1; off >>= 1) {
      const float ov = __shfl_xor(v, off, 32);
      const int   oi = __shfl_xor(id, off, 32);
      if (ov < v || (ov == v && oi < id)) { v = ov; id = oi; }
    }
    if (c == 0) {  // lanes 0 and 16 publish rows m0+r and m0+8+r
      const int row = l0 + m0 + half * 8 + r;
      const unsigned long long key =
          ((unsigned long long)orderedKey(v) << 32) | (unsigned int)id;
      atomicMin(&packed[(size_t)b * L_ + row], key);
    }
  }
}

// ---------------------------------------------------------------- finalize
__global__ void tsel_final_kernel(const unsigned long long* __restrict__ packed,
                                  float* __restrict__ gids,
                                  float* __restrict__ gsizes) {
  const int tid = blockIdx.x * blockDim.x + threadIdx.x;
  if (tid >= B_ * L_) return;
  const int b = tid / L_;
  const unsigned int idx = (unsigned int)(packed[tid] & 0xFFFFFFFFu);
  gids[tid] = (float)idx;
  atomicAdd(&gsizes[b * K_ + idx], 1.0f);
}

// ---------------------------------------------------------------- launch
extern "C" void kernel_launch(void* const* d_in, const int* in_sizes, int n_in,
                              void* d_out, int out_size, void* d_ws, size_t ws_size,
                              hipStream_t stream) {
  const float* x = (const float*)d_in[0];
  // d_in[1] (W) and d_in[2] (b) provably do not affect the outputs:
  // saliency == 1.0 for every token, so top_k indices == arange(k_target).
  (void)in_sizes; (void)n_in; (void)out_size; (void)ws_size;

  float* out    = (float*)d_out;
  float* xsel   = out;                                   // [B, K, D]
  float* gids   = xsel + (size_t)B_ * K_ * D_;           // [B, L]
  float* gsizes = gids + (size_t)B_ * L_;                // [B, K]

  unsigned long long* packed = (unsigned long long*)d_ws;            // [B*L] keys
  float* s2 = (float*)((char*)d_ws + (size_t)B_ * L_ * sizeof(unsigned long long));

  tsel_init_kernel<<<(B_ * L_ + 255) / 256, 256, 0, stream>>>(packed, gsizes);
  tsel_s2_copy_kernel<<<(B_ * K_) / 8, 256, 0, stream>>>(x, xsel, s2);
  dim3 grid(K_ / TK, L_ / TL, B_);
  tsel_argmin_kernel<<<grid, 256, 0, stream>>>(x, s2, packed);
  tsel_final_kernel<<<(B_ * L_ + 255) / 256, 256, 0, stream>>>(packed, gids, gsizes);
}